// RoiPoolingConv_28011776704910
// MI455X (gfx1250) — compile-verified
//
#include <hip/hip_runtime.h>

// RoI max pooling, MI455X (gfx1250, wave32).
// One 64-thread block (2 waves) per (roi, bin). Each thread owns 4 channels
// (float4 => 16B), so one patch pixel = 64 lanes * 16B = 1KB fully coalesced.
// Patch (<=5x5 pixels, <=25KB) is staged into LDS with CDNA5 async
// global->LDS loads (ASYNCcnt), then max-reduced with ds_load_b128.

#define POOLP 7
#define HH    64
#define WW    64
#define CC    256
#define MAXD  5   // ceil(w/7) <= 5 since w,h in [7,32)

typedef float f4 __attribute__((ext_vector_type(4)));
typedef int   i4v __attribute__((ext_vector_type(4)));

#define AS1 __attribute__((address_space(1)))
#define AS3 __attribute__((address_space(3)))

#if defined(__gfx1250__) &&                                                   \
    __has_builtin(__builtin_amdgcn_global_load_async_to_lds_b128) &&          \
    __has_builtin(__builtin_amdgcn_s_wait_asynccnt)
#define USE_ASYNC_LDS 1
#else
#define USE_ASYNC_LDS 0
#endif

__global__ __launch_bounds__(64) void roi_pool_gfx1250(
    const float* __restrict__ img,   // [64,64,256]
    const float* __restrict__ rois,  // [R,4] (x,y,w,h)
    float* __restrict__ out,         // [R,7,7,256]
    int nbins)                       // R*49
{
    __shared__ __align__(16) float smem[MAXD * MAXD * CC];  // 25 KB

    const int bid = blockIdx.x;
    if (bid >= nbins) return;
    const int tid = threadIdx.x;          // 0..63, owns channels 4*tid..4*tid+3
    const int r  = bid / (POOLP * POOLP);
    const int ab = bid % (POOLP * POOLP);
    const int a  = ab / POOLP;            // jy bin: cols via y/h
    const int b  = ab % POOLP;            // ix bin: rows via x/w

    const float x = rois[4 * r + 0];
    const float y = rois[4 * r + 1];
    const float w = rois[4 * r + 2];
    const float h = rois[4 * r + 3];

    // Match JAX's separately-rounded fp32 ops (no FMA contraction).
    const float rl  = __fdiv_rn(w, (float)POOLP);
    const float cl  = __fdiv_rn(h, (float)POOLP);
    const float x1f = __fadd_rn(x, __fmul_rn((float)b, rl));
    const float y1f = __fadd_rn(y, __fmul_rn((float)a, cl));
    const int x1 = (int)x1f;
    const int x2 = (int)__fadd_rn(x1f, rl);
    const int y1 = (int)y1f;
    const int y2 = (int)__fadd_rn(y1f, cl);
    const int dx = min(max(1, x2 - x1), MAXD);  // row count
    const int dy = min(max(1, y2 - y1), MAXD);  // col count

    f4 acc;
    acc[0] = -__builtin_inff(); acc[1] = -__builtin_inff();
    acc[2] = -__builtin_inff(); acc[3] = -__builtin_inff();

#if USE_ASYNC_LDS
    // ---- stage patch into LDS with async global->LDS b128 loads ----
    int p = 0;
    for (int d2 = 0; d2 < dx; ++d2) {
        const int row = min(max(x1 + d2, 0), HH - 1);
        const float* rowbase = img + (size_t)(row * WW) * CC;
        for (int d1 = 0; d1 < dy; ++d1) {
            const int col = min(max(y1 + d1, 0), WW - 1);
            const float* gsrc = rowbase + (size_t)col * CC + 4 * tid;
            __builtin_amdgcn_global_load_async_to_lds_b128(
                (AS1 i4v*)gsrc,
                (AS3 i4v*)&smem[p * CC + 4 * tid],
                0, 0);
            ++p;
        }
    }
    __builtin_amdgcn_s_wait_asynccnt(0);   // lane reads only what it staged

    // ---- max-reduce from LDS (ds_load_b128) ----
    for (int q = 0; q < p; ++q) {
        const f4 v = *(const f4*)&smem[q * CC + 4 * tid];
        acc[0] = fmaxf(acc[0], v[0]);
        acc[1] = fmaxf(acc[1], v[1]);
        acc[2] = fmaxf(acc[2], v[2]);
        acc[3] = fmaxf(acc[3], v[3]);
    }
#else
    // ---- fallback: direct coalesced global_load_b128 gather ----
    (void)smem;
    for (int d2 = 0; d2 < dx; ++d2) {
        const int row = min(max(x1 + d2, 0), HH - 1);
        const float* rowbase = img + (size_t)(row * WW) * CC;
        for (int d1 = 0; d1 < dy; ++d1) {
            const int col = min(max(y1 + d1, 0), WW - 1);
            const f4 v = *(const f4*)(rowbase + (size_t)col * CC + 4 * tid);
            acc[0] = fmaxf(acc[0], v[0]);
            acc[1] = fmaxf(acc[1], v[1]);
            acc[2] = fmaxf(acc[2], v[2]);
            acc[3] = fmaxf(acc[3], v[3]);
        }
    }
#endif

    // out[((r*7 + a)*7 + b)*256 + c] == out[bid*256 + c]
    *(f4*)(out + (size_t)bid * CC + 4 * tid) = acc;
}

extern "C" void kernel_launch(void* const* d_in, const int* in_sizes, int n_in,
                              void* d_out, int out_size, void* d_ws, size_t ws_size,
                              hipStream_t stream) {
    const float* img  = (const float*)d_in[0];
    const float* rois = (const float*)d_in[1];
    float* out = (float*)d_out;

    const int R = in_sizes[1] / 4;           // rois is [R,4]
    const int nbins = R * POOLP * POOLP;     // one block per (roi, bin)

    dim3 grid(nbins);
    dim3 block(64);                          // 2 waves (wave32)
    hipLaunchKernelGGL(roi_pool_gfx1250, grid, block, 0, stream,
                       img, rois, out, nbins);
}